// Network_42803644072809
// MI455X (gfx1250) — compile-verified
//
#include <hip/hip_runtime.h>
#include <hip/hip_bf16.h>
#include <math.h>

typedef float v2f __attribute__((ext_vector_type(2)));
typedef float v4f __attribute__((ext_vector_type(4)));
typedef float v8f __attribute__((ext_vector_type(8)));

#define EPSF     1e-8f
#define RAD2DEG  57.29577951308232f

__device__ __forceinline__ float angle_deg(float q0, float q1, float q2,
                                           float l0, float l1, float l2) {
    const float np  = sqrtf(q0 * q0 + q1 * q1 + q2 * q2);
    const float inv = 1.0f / fmaxf(np, EPSF);
    q0 *= inv; q1 *= inv; q2 *= inv;
    const float na = fmaxf(sqrtf(q0 * q0 + q1 * q1 + q2 * q2), EPSF);
    const float nb = fmaxf(sqrtf(l0 * l0 + l1 * l1 + l2 * l2), EPSF);
    const float cs = (q0 * l0 + q1 * l1 + q2 * l2) / (na * nb);
    float ang = acosf(cs) * RAD2DEG;     // |cs|>1 -> NaN, matches reference
    if (isnan(ang)) ang = 0.0f;
    return ang;
}

// Persistent waves grid-stride over FULL 32-row tiles (two 16-row WMMA tiles,
// K=64 in K-steps of 4). K order permuted so each lane loads contiguous float4:
//   fragment 2k   : lanes 0-15 -> cols 8k+0,8k+1 ; lanes 16-31 -> cols 8k+4,8k+5
//   fragment 2k+1 : lanes 0-15 -> cols 8k+2,8k+3 ; lanes 16-31 -> cols 8k+6,8k+7
// B fragments (built once per wave) use the identical permutation -> exact GEMM.
// Tail rows (N % 32) are handled by one wave with plain VALU (no WMMA).
__global__ __launch_bounds__(256) void Network_42803644072809_kernel(
    const float* __restrict__ data,   // [N,64]
    const float* __restrict__ lab,    // [N,3]
    const float* __restrict__ W,      // [64,3]
    const float* __restrict__ bias,   // [3]
    float* __restrict__ out,          // [N]
    int N)
{
    __shared__ float lds[8][32][4];   // per-wave 32 rows x (3 pred cols, padded)

    const int lane  = threadIdx.x & 31;
    const int wave  = threadIdx.x >> 5;
    const long nFull  = (long)N >> 5;                 // full 32-row tiles
    const long nWaves = (long)gridDim.x * 8;
    const long wave0  = (long)blockIdx.x * 8 + wave;

    const int n     = lane & 15;              // B/C column this lane holds
    const int khalf = lane >> 4;              // 4-col group within 8-col super-step
    const int r     = lane & 15;              // A-matrix row this lane holds
    const int nc    = (n < 3) ? n : 2;        // clamped col for branch-free W loads

    // ---- Hoisted: bias and B fragments (branch-free select) ----
    const float b0 = bias[0], b1 = bias[1], b2 = bias[2];

    v2f bf[16];
#pragma unroll
    for (int k8 = 0; k8 < 8; ++k8) {
        const int kb = 8 * k8 + 4 * khalf;
        float x0 = W[(kb + 0) * 3 + nc];
        float y0 = W[(kb + 1) * 3 + nc];
        float x1 = W[(kb + 2) * 3 + nc];
        float y1 = W[(kb + 3) * 3 + nc];
        if (n >= 3) { x0 = 0.f; y0 = 0.f; x1 = 0.f; y1 = 0.f; }  // cndmask
        bf[2 * k8].x     = x0;  bf[2 * k8].y     = y0;
        bf[2 * k8 + 1].x = x1;  bf[2 * k8 + 1].y = y1;
    }

    // ---- Hot loop: pure pointer increments, no clamping ----
    const float* pA0 = data + (wave0 * 32 + r) * 64 + 4 * khalf;
    const float* pA1 = pA0 + 16 * 64;
    const float* pL  = lab + (wave0 * 32 + lane) * 3;
    float*       pO  = out +  wave0 * 32 + lane;
    const long stepA = nWaves * 32 * 64;
    const long stepL = nWaves * 32 * 3;
    const long stepO = nWaves * 32;

    for (long tile = wave0; tile < nFull; tile += nWaves) {
        // Stage ALL A fragments (16 x b128 NT) + lab, as one batched clause.
        v4f a0[8], a1[8];
#pragma unroll
        for (int k8 = 0; k8 < 8; ++k8) {
            a0[k8] = __builtin_nontemporal_load((const v4f*)(pA0 + 8 * k8));
            a1[k8] = __builtin_nontemporal_load((const v4f*)(pA1 + 8 * k8));
        }
        const float l0 = __builtin_nontemporal_load(pL + 0);
        const float l1 = __builtin_nontemporal_load(pL + 1);
        const float l2 = __builtin_nontemporal_load(pL + 2);

        // Keep the load stage intact: nothing moves across this point.
        __builtin_amdgcn_sched_barrier(0);

        // WMMA chain: 32 x v_wmma_f32_16x16x4_f32, interleaved accumulators.
        v8f c0 = {0.f,0.f,0.f,0.f,0.f,0.f,0.f,0.f};
        v8f c1 = c0;
#pragma unroll
        for (int k8 = 0; k8 < 8; ++k8) {
            v2f x0; x0.x = a0[k8].x; x0.y = a0[k8].y;   // K pair (kb+0, kb+1)
            v2f x1; x1.x = a0[k8].z; x1.y = a0[k8].w;   // K pair (kb+2, kb+3)
            v2f y0; y0.x = a1[k8].x; y0.y = a1[k8].y;
            v2f y1; y1.x = a1[k8].z; y1.y = a1[k8].w;
            c0 = __builtin_amdgcn_wmma_f32_16x16x4_f32(false, x0, false, bf[2*k8],
                                                       (short)0, c0, false, false);
            c1 = __builtin_amdgcn_wmma_f32_16x16x4_f32(false, y0, false, bf[2*k8],
                                                       (short)0, c1, false, false);
            c0 = __builtin_amdgcn_wmma_f32_16x16x4_f32(false, x1, false, bf[2*k8+1],
                                                       (short)0, c0, false, false);
            c1 = __builtin_amdgcn_wmma_f32_16x16x4_f32(false, y1, false, bf[2*k8+1],
                                                       (short)0, c1, false, false);
        }

        // Transpose pred cols 0..2 through LDS.
        // C layout: lane holds C[M = j + 8*khalf][N = n] in c[j].
        if (n < 3) {
#pragma unroll
            for (int j = 0; j < 8; ++j) {
                lds[wave][j + 8 * khalf][n]      = c0[j];
                lds[wave][16 + j + 8 * khalf][n] = c1[j];
            }
        }
        asm volatile("s_wait_dscnt 0" ::: "memory");   // same-wave LDS RAW

        // Epilogue: lane L owns local row L.
        const float q0 = lds[wave][lane][0] + b0;
        const float q1 = lds[wave][lane][1] + b1;
        const float q2 = lds[wave][lane][2] + b2;
        const float ang = angle_deg(q0, q1, q2, l0, l1, l2);

        __builtin_nontemporal_store(ang, pO);

        pA0 += stepA; pA1 += stepA; pL += stepL; pO += stepO;
    }

    // ---- Tail (N % 32 rows): one wave, plain VALU, no WMMA ----
    const long tail = (long)N - (nFull << 5);
    if (tail > 0 && blockIdx.x == 0 && wave == 0) {
        const long g = (nFull << 5) + lane;
        if (lane < tail) {
            float q0 = b0, q1 = b1, q2 = b2;
            const float* row = data + g * 64;
#pragma unroll 4
            for (int k = 0; k < 64; ++k) {
                const float d = row[k];
                q0 += d * W[k * 3 + 0];
                q1 += d * W[k * 3 + 1];
                q2 += d * W[k * 3 + 2];
            }
            out[g] = angle_deg(q0, q1, q2,
                               lab[g * 3 + 0], lab[g * 3 + 1], lab[g * 3 + 2]);
        }
    }
}

extern "C" void kernel_launch(void* const* d_in, const int* in_sizes, int n_in,
                              void* d_out, int out_size, void* d_ws, size_t ws_size,
                              hipStream_t stream) {
    const float* data = (const float*)d_in[0];   // [N,64]
    const float* lab  = (const float*)d_in[1];   // [N,3]
    const float* W    = (const float*)d_in[2];   // [64,3]
    const float* bias = (const float*)d_in[3];   // [3]
    float* out = (float*)d_out;

    const int  N      = in_sizes[0] / 64;
    const long nTiles = ((long)N + 31) / 32;          // 32 rows per wave-tile
    long blocks = (nTiles + 7) / 8;                   // at most one tile per wave
    if (blocks > 1024) blocks = 1024;                 // persistent: ~8 tiles/wave
    if (blocks < 1)    blocks = 1;

    Network_42803644072809_kernel<<<(int)blocks, 256, 0, stream>>>(
        data, lab, W, bias, out, N);
}